// STGCN_12455405158470
// MI455X (gfx1250) — compile-verified
//
#include <hip/hip_runtime.h>

#define TT 8
#define T1 (TT + 6)   // 14  g0 times  (t = tbase-3+i)
#define T2 (TT + 4)   // 12  h0 times  (t = tbase-2+i)
#define T3 (TT + 2)   // 10  g1 times  (t = tbase-1+i)
#define NB 64
#define NTIME 512
#define NV 21
#define NC 32
#define NCIN 2
#define NOUT 64

typedef __attribute__((ext_vector_type(2))) float v2f;
typedef __attribute__((ext_vector_type(8))) float v8f;

__device__ __forceinline__ v8f wmma4(v2f a, v2f b, v8f c) {
  // V_WMMA_F32_16X16X4_F32 : D = A(16x4) * B(4x16) + C
  return __builtin_amdgcn_wmma_f32_16x16x4_f32(
      /*neg_a=*/false, a, /*neg_b=*/false, b,
      /*c_mod=*/(short)0, c, /*reuse_a=*/false, /*reuse_b=*/false);
}

// Hand-skeleton graph (compile-time constant in the reference)
__device__ const unsigned char g_pi[20] = {0,1,2,3,0,5,6,7,0,9,10,11,0,13,14,15,0,17,18,19};
__device__ const unsigned char g_pj[20] = {1,2,3,4,5,6,7,8,9,10,11,12,13,14,15,16,17,18,19,20};
__device__ const float g_dinv[21] = {
  0.4082482905f,
  0.5773502692f,0.5773502692f,0.5773502692f,0.7071067812f,
  0.5773502692f,0.5773502692f,0.5773502692f,0.7071067812f,
  0.5773502692f,0.5773502692f,0.5773502692f,0.7071067812f,
  0.5773502692f,0.5773502692f,0.5773502692f,0.7071067812f,
  0.5773502692f,0.5773502692f,0.5773502692f,0.7071067812f};

__global__ __launch_bounds__(256) void stgcn_fused(
    const float* __restrict__ x,
    const float* __restrict__ gcn_w0, const float* __restrict__ gcn_b0,
    const float* __restrict__ tcn_w0, const float* __restrict__ tcn_b0,
    const float* __restrict__ bn_g0, const float* __restrict__ bn_b0,
    const float* __restrict__ bn_m0, const float* __restrict__ bn_v0,
    const float* __restrict__ res_w, const float* __restrict__ res_b,
    const float* __restrict__ res_bn_g, const float* __restrict__ res_bn_b,
    const float* __restrict__ res_bn_m, const float* __restrict__ res_bn_v,
    const float* __restrict__ gcn_w1, const float* __restrict__ gcn_b1,
    const float* __restrict__ tcn_w1, const float* __restrict__ tcn_b1,
    const float* __restrict__ bn_g1, const float* __restrict__ bn_b1,
    const float* __restrict__ bn_m1, const float* __restrict__ bn_v1,
    const float* __restrict__ fc_w, const float* __restrict__ fc_b,
    float* __restrict__ out)
{
  __shared__ float sA[NV * NV];              // 441
  __shared__ float s_in[T1 * NV * NCIN];     // 588   raw x tile (also residual src)
  __shared__ float s_xa[T1 * NV * NCIN];     // 588   graph-agg of x
  __shared__ float s_tw0[96 * NC];           // 3072  tcn0 weights, [kappa=k*32+ci][co]
  __shared__ float s_tw1[96 * NC];           // 3072
  __shared__ float s_gw1[NC * NC];           // 1024  [co][ci]
  __shared__ float s_g0[T1 * NV * NC];       // 9408  [i][v][c]; reused: hagg, then fc partials
  __shared__ float s_h0[T2 * NV * NC];       // 8064  [i][v][c]; h1 written in place
  __shared__ float s_g1[T3 * NV * NC];       // 6720  [i][v][c]
  __shared__ float s_gw0[NC * NCIN];
  __shared__ float s_gb0[NC], s_gb1[NC];
  __shared__ float s_s0[NC], s_o0[NC], s_s1[NC], s_o1[NC];
  __shared__ float s_rA[NC], s_rB[NC], s_rC[NC];
  __shared__ float s_fcb[NOUT];

  const int tid  = threadIdx.x;
  const int nthr = 256;
  const int lane = tid & 31;
  const int wave = tid >> 5;
  const int row  = lane & 15;   // A-row / B-col / D-col
  const int hi   = lane >> 4;   // K-pair / D-row-half selector

  const int bid   = blockIdx.x;
  const int tile  = bid & 63;          // NTIME/TT = 64 tiles
  const int hand  = (bid >> 6) & 1;
  const int b     = bid >> 7;
  const int tbase = tile * TT;

  // ---- build normalized adjacency A = D^-1/2 (Adj+I) D^-1/2 ----
  for (int idx = tid; idx < NV * NV; idx += nthr) {
    int i = idx / NV, j = idx % NV;
    bool conn = (i == j);
    for (int p = 0; p < 20; ++p)
      conn = conn || ((int)g_pi[p] == i && (int)g_pj[p] == j) ||
                     ((int)g_pi[p] == j && (int)g_pj[p] == i);
    sA[idx] = conn ? g_dinv[i] * g_dinv[j] : 0.0f;
  }
  // ---- stage weights / folded BN params into LDS ----
  for (int idx = tid; idx < 96 * NC; idx += nthr) {
    int co = idx & 31, kap = idx >> 5, k = kap >> 5, ci = kap & 31;
    s_tw0[idx] = tcn_w0[(co * NC + ci) * 3 + k];
    s_tw1[idx] = tcn_w1[(co * NC + ci) * 3 + k];
  }
  for (int idx = tid; idx < NC * NC; idx += nthr) s_gw1[idx] = gcn_w1[idx];
  if (tid < NC * NCIN) s_gw0[tid] = gcn_w0[tid];
  if (tid < NC) {
    s_gb0[tid] = gcn_b0[tid];
    s_gb1[tid] = gcn_b1[tid];
    float sc0 = bn_g0[tid] * rsqrtf(bn_v0[tid] + 1e-5f);
    s_s0[tid] = sc0;
    s_o0[tid] = (tcn_b0[tid] - bn_m0[tid]) * sc0 + bn_b0[tid];
    float sc1 = bn_g1[tid] * rsqrtf(bn_v1[tid] + 1e-5f);
    s_s1[tid] = sc1;
    s_o1[tid] = (tcn_b1[tid] - bn_m1[tid]) * sc1 + bn_b1[tid];
    float rsc = res_bn_g[tid] * rsqrtf(res_bn_v[tid] + 1e-5f);
    s_rA[tid] = res_w[tid * 2 + 0] * rsc;
    s_rB[tid] = res_w[tid * 2 + 1] * rsc;
    s_rC[tid] = (res_b[tid] - res_bn_m[tid]) * rsc + res_bn_b[tid];
  }
  if (tid < NOUT) s_fcb[tid] = fc_b[tid];

  // ---- load x tile (zero outside [0,T)) ----
  for (int idx = tid; idx < T1 * NV * NCIN; idx += nthr) {
    int i = idx / (NV * NCIN), r = idx % (NV * NCIN);
    int v = r >> 1, ci = r & 1;
    int t = tbase - 3 + i;
    float val = 0.0f;
    if (t >= 0 && t < NTIME)
      val = x[(((b * NTIME + t) * 2 + hand) * NV + v) * NCIN + ci];
    s_in[idx] = val;
  }
  __syncthreads();

  // ---- graph aggregate of x (V=21, VALU) ----
  for (int idx = tid; idx < T1 * NV * NCIN; idx += nthr) {
    int i = idx / (NV * NCIN), r = idx % (NV * NCIN);
    int v = r >> 1, ci = r & 1;
    float acc = 0.0f;
    const float* Ar = &sA[v * NV];
    const float* xr = &s_in[i * NV * NCIN + ci];
    for (int u = 0; u < NV; ++u) acc += Ar[u] * xr[u * NCIN];
    s_xa[idx] = acc;
  }
  __syncthreads();

  // ---- g0 = gcn0 (2->32) ; rows at invalid t zeroed (conv zero-pad semantics)
  for (int idx = tid; idx < T1 * NV * NC; idx += nthr) {
    int c = idx & 31, rv = idx >> 5;
    int v = rv % NV, i = rv / NV;
    int t = tbase - 3 + i;
    float val = 0.0f;
    if (t >= 0 && t < NTIME) {
      float xa0 = s_xa[(i * NV + v) * 2 + 0];
      float xa1 = s_xa[(i * NV + v) * 2 + 1];
      val = s_gw0[c * 2 + 0] * xa0 + s_gw0[c * 2 + 1] * xa1 + s_gb0[c];
    }
    s_g0[idx] = val;
  }
  __syncthreads();

  // ---- TCN0 (K=96) + BN0 + conv-residual + relu -> h0 : WMMA f32 16x16x4 ----
  // Out-of-range A rows (clamped to row 0) only pollute out-of-range D rows,
  // which are discarded by the store guard -> no mask multiply needed.
  {
    const int M = T2 * NV;  // 252
    for (int tIdx = wave; tIdx < 16 * 2; tIdx += 8) {
      int mt = tIdx >> 1, nt = tIdx & 1;
      v8f acc = {0.f,0.f,0.f,0.f,0.f,0.f,0.f,0.f};
      int mA = mt * 16 + row;
      int mC = (mA < M) ? mA : 0;
      int t2 = mC / NV, vv = mC % NV;
      int colN = nt * 16 + row;
      for (int ks = 0; ks < 24; ++ks) {
        int k0 = ks * 4 + hi * 2;
        v2f a, bb;
        a.x = s_g0[(t2 + (k0 >> 5)) * (NV * NC) + vv * NC + (k0 & 31)];
        a.y = s_g0[(t2 + ((k0 + 1) >> 5)) * (NV * NC) + vv * NC + ((k0 + 1) & 31)];
        bb.x = s_tw0[k0 * NC + colN];
        bb.y = s_tw0[(k0 + 1) * NC + colN];
        acc = wmma4(a, bb, acc);
      }
#pragma unroll
      for (int j = 0; j < 8; ++j) {
        int m = mt * 16 + j + hi * 8;
        if (m < M) {
          int te = m / NV, ve = m % NV;
          int co = colN;
          float val = acc[j] * s_s0[co] + s_o0[co];
          float x0 = s_in[((te + 1) * NV + ve) * 2 + 0];
          float x1 = s_in[((te + 1) * NV + ve) * 2 + 1];
          val += s_rA[co] * x0 + s_rB[co] * x1 + s_rC[co];
          s_h0[(te * NV + ve) * NC + co] = fmaxf(val, 0.0f);
        }
      }
    }
  }
  __syncthreads();

  // ---- graph aggregate of h0 (reuse s_g0 storage) ----
  float* s_hagg = s_g0;
  for (int idx = tid; idx < T3 * NV * NC; idx += nthr) {
    int ci = idx & 31, rv = idx >> 5;
    int v = rv % NV, t3 = rv / NV;
    float acc = 0.0f;
    const float* Ar = &sA[v * NV];
    const float* hr = &s_h0[(t3 + 1) * (NV * NC) + ci];
    for (int u = 0; u < NV; ++u) acc += Ar[u] * hr[u * NC];
    s_hagg[idx] = acc;
  }
  __syncthreads();

  // ---- GCN1 (K=32) -> g1 (+bias, invalid-t rows zeroed) : WMMA ----
  {
    const int M = T3 * NV;  // 210
    for (int tIdx = wave; tIdx < 14 * 2; tIdx += 8) {
      int mt = tIdx >> 1, nt = tIdx & 1;
      v8f acc = {0.f,0.f,0.f,0.f,0.f,0.f,0.f,0.f};
      int mA = mt * 16 + row;
      int mC = (mA < M) ? mA : 0;
      int colN = nt * 16 + row;
      for (int ks = 0; ks < 8; ++ks) {
        int k0 = ks * 4 + hi * 2;
        v2f a, bb;
        a.x = s_hagg[mC * NC + k0];
        a.y = s_hagg[mC * NC + k0 + 1];
        bb.x = s_gw1[colN * NC + k0];      // B[ci][co] = gcn_w1[co][ci]
        bb.y = s_gw1[colN * NC + k0 + 1];
        acc = wmma4(a, bb, acc);
      }
#pragma unroll
      for (int j = 0; j < 8; ++j) {
        int m = mt * 16 + j + hi * 8;
        if (m < M) {
          int t3 = m / NV;
          int t = tbase - 1 + t3;
          float val = acc[j] + s_gb1[colN];
          s_g1[m * NC + colN] = (t >= 0 && t < NTIME) ? val : 0.0f;
        }
      }
    }
  }
  __syncthreads();

  // ---- TCN1 (K=96) + BN1 + identity residual + relu -> h1 (in place) : WMMA ----
  {
    const int M = TT * NV;  // 168
    for (int tIdx = wave; tIdx < 11 * 2; tIdx += 8) {
      int mt = tIdx >> 1, nt = tIdx & 1;
      v8f acc = {0.f,0.f,0.f,0.f,0.f,0.f,0.f,0.f};
      int mA = mt * 16 + row;
      int mC = (mA < M) ? mA : 0;
      int t4 = mC / NV, vv = mC % NV;
      int colN = nt * 16 + row;
      for (int ks = 0; ks < 24; ++ks) {
        int k0 = ks * 4 + hi * 2;
        v2f a, bb;
        a.x = s_g1[((t4 + (k0 >> 5)) * NV + vv) * NC + (k0 & 31)];
        a.y = s_g1[((t4 + ((k0 + 1) >> 5)) * NV + vv) * NC + ((k0 + 1) & 31)];
        bb.x = s_tw1[k0 * NC + colN];
        bb.y = s_tw1[(k0 + 1) * NC + colN];
        acc = wmma4(a, bb, acc);
      }
#pragma unroll
      for (int j = 0; j < 8; ++j) {
        int m = mt * 16 + j + hi * 8;
        if (m < M) {
          int te = m / NV, ve = m % NV;
          int co = colN;
          int hidx = ((te + 2) * NV + ve) * NC + co;  // h1 t -> h0 slot t+2
          float val = acc[j] * s_s1[co] + s_o1[co] + s_h0[hidx];
          s_h0[hidx] = fmaxf(val, 0.0f);  // element-wise in-place: safe
        }
      }
    }
  }
  __syncthreads();

  // ---- FC 672->64 : WMMA, K=672 split across all 8 waves (84 steps each) ----
  // waves 0..3: nt = wave,   ks in [0,84)
  // waves 4..7: nt = wave-4, ks in [84,168)  -> partials via LDS (reuse s_g0)
  {
    float* s_red = s_g0;  // 4 tiles * 256 floats = 1024, fits easily
    int nt  = wave & 3;
    int ksb = (wave >> 2) * 84;
    v8f acc = {0.f,0.f,0.f,0.f,0.f,0.f,0.f,0.f};
    int t4 = (row < TT) ? row : (TT - 1);   // clamp; rows 8..15 discarded
    int colN = nt * 16 + row;               // output feature o
    const float* h1base = &s_h0[(t4 + 2) * (NV * NC)];
    for (int ks = ksb; ks < ksb + 84; ++ks) {
      int f0 = ks * 4 + hi * 2;
      int f1 = f0 + 1;
      int c0 = f0 / NV, v0 = f0 % NV;
      int c1 = f1 / NV, v1 = f1 % NV;
      v2f a, bb;
      a.x = h1base[v0 * NC + c0];
      a.y = h1base[v1 * NC + c1];
      bb.x = fc_w[colN * 672 + f0];
      bb.y = fc_w[colN * 672 + f1];
      acc = wmma4(a, bb, acc);
    }
    if (wave >= 4) {
#pragma unroll
      for (int j = 0; j < 8; ++j) s_red[nt * 256 + lane * 8 + j] = acc[j];
    }
    __syncthreads();
    if (wave < 4) {
#pragma unroll
      for (int j = 0; j < 8; ++j) {
        int m = j + hi * 8;
        if (m < TT) {
          int t = tbase + m;
          float val = acc[j] + s_red[nt * 256 + lane * 8 + j] + s_fcb[colN];
          out[(((size_t)b * NTIME + t) * 2 + hand) * NOUT + colN] = val;
        }
      }
    }
  }
}

extern "C" void kernel_launch(void* const* d_in, const int* in_sizes, int n_in,
                              void* d_out, int out_size, void* d_ws, size_t ws_size,
                              hipStream_t stream) {
  (void)in_sizes; (void)n_in; (void)d_ws; (void)ws_size; (void)out_size;
  const float* x        = (const float*)d_in[0];
  const float* gcn_w0   = (const float*)d_in[1];
  const float* gcn_b0   = (const float*)d_in[2];
  const float* tcn_w0   = (const float*)d_in[3];
  const float* tcn_b0   = (const float*)d_in[4];
  const float* bn_g0    = (const float*)d_in[5];
  const float* bn_b0    = (const float*)d_in[6];
  const float* bn_m0    = (const float*)d_in[7];
  const float* bn_v0    = (const float*)d_in[8];
  const float* res_w    = (const float*)d_in[9];
  const float* res_b    = (const float*)d_in[10];
  const float* res_bn_g = (const float*)d_in[11];
  const float* res_bn_b = (const float*)d_in[12];
  const float* res_bn_m = (const float*)d_in[13];
  const float* res_bn_v = (const float*)d_in[14];
  const float* gcn_w1   = (const float*)d_in[15];
  const float* gcn_b1   = (const float*)d_in[16];
  const float* tcn_w1   = (const float*)d_in[17];
  const float* tcn_b1   = (const float*)d_in[18];
  const float* bn_g1    = (const float*)d_in[19];
  const float* bn_b1    = (const float*)d_in[20];
  const float* bn_m1    = (const float*)d_in[21];
  const float* bn_v1    = (const float*)d_in[22];
  const float* fc_w     = (const float*)d_in[23];
  const float* fc_b     = (const float*)d_in[24];

  dim3 grid(NB * 2 * (NTIME / TT));  // 8192 workgroups
  dim3 block(256);
  stgcn_fused<<<grid, block, 0, stream>>>(
      x, gcn_w0, gcn_b0, tcn_w0, tcn_b0, bn_g0, bn_b0, bn_m0, bn_v0,
      res_w, res_b, res_bn_g, res_bn_b, res_bn_m, res_bn_v,
      gcn_w1, gcn_b1, tcn_w1, tcn_b1, bn_g1, bn_b1, bn_m1, bn_v1,
      fc_w, fc_b, (float*)d_out);
}